// EquivariantLayerNorm_89833535963952
// MI455X (gfx1250) — compile-verified
//
#include <hip/hip_runtime.h>

// ---------------------------------------------------------------------------
// EquivariantLayerNorm for irreps 128x0e + 64x1o + 32x2e  (DIM = 480, N = 100k)
//
// Memory-bound: 384 MB total traffic -> ~16.5 us floor @ 23.3 TB/s.
// Compute is ~0.24 GFLOP total (negligible). WMMA cannot help a per-node
// reduction/normalize stream; the CDNA5 paths that matter are wave32,
// coalesced b128 vmem, and the ASYNCcnt global->LDS staging pipe.
// ---------------------------------------------------------------------------

#define ELN_DIM        480
#define ELN_CHUNKS     120   // float4 chunks per node
#define ELN_NSCALAR    128
#define ELN_EPS        1e-5f
#define WAVES_PER_BLK  8

// Builtin pointee type per hipcc diagnostic: int __attribute__((vector_size(16)))
typedef int eln_v4i __attribute__((vector_size(4 * sizeof(int))));

// ---- gfx1250 async global->LDS copy (ASYNCcnt path), with asm fallback ----
#if defined(__gfx1250__) && __has_builtin(__builtin_amdgcn_global_load_async_to_lds_b128)
#define ELN_ASYNC_B128(gp, lp)                                                  \
    __builtin_amdgcn_global_load_async_to_lds_b128(                             \
        (__attribute__((address_space(1))) eln_v4i*)(gp),                       \
        (__attribute__((address_space(3))) eln_v4i*)(lp), 0, 0)
#else
#define ELN_ASYNC_B128(gp, lp)                                                  \
    asm volatile("global_load_async_to_lds_b128 %0, %1, off"                    \
                 :: "v"((unsigned)(unsigned long long)(lp)),                    \
                    "v"((unsigned long long)(gp))                               \
                 : "memory")
#endif

#if defined(__gfx1250__) && __has_builtin(__builtin_amdgcn_s_wait_asynccnt)
#define ELN_WAIT_ASYNC() __builtin_amdgcn_s_wait_asynccnt(0)
#else
#define ELN_WAIT_ASYNC() asm volatile("s_wait_asynccnt 0" ::: "memory")
#endif

__global__ __launch_bounds__(256)
void eln_kernel(const float* __restrict__ in,
                const float* __restrict__ aw,   // [224] per-multiplicity weight
                const float* __restrict__ ab,   // [128] scalar bias
                float* __restrict__ out,
                int nNodes)
{
    __shared__ __align__(16) float  wexp[ELN_DIM];      // per-element weight
    __shared__ __align__(16) float  bexp[ELN_NSCALAR];  // per-element bias
    __shared__ __align__(16) float4 stage[WAVES_PER_BLK][ELN_CHUNKS];

    const int tid  = threadIdx.x;
    const int wid  = tid >> 5;
    const int lane = tid & 31;

    // Expand per-multiplicity affine weight to per-element (once per block).
    for (int i = tid; i < ELN_DIM; i += 256) {
        float w;
        if (i < 128)       w = aw[i];                       // l=0: d=1
        else if (i < 320)  w = aw[128 + (i - 128) / 3];     // l=1: d=3
        else               w = aw[192 + (i - 320) / 5];     // l=2: d=5
        wexp[i] = w;
    }
    for (int i = tid; i < ELN_NSCALAR; i += 256) bexp[i] = ab[i];
    __syncthreads();

    const int node = blockIdx.x * WAVES_PER_BLK + wid;
    if (node >= nNodes) return;

    const float4* __restrict__ src = (const float4*)in + (size_t)node * ELN_CHUNKS;
    float4*       __restrict__ dst = (float4*)out       + (size_t)node * ELN_CHUNKS;

    // ---- Stage 1: async-copy this node's 1920 B into LDS (coalesced b128) --
#pragma unroll
    for (int j = 0; j < 4; ++j) {
        const int c = lane + 32 * j;
        if (c < ELN_CHUNKS) {
            ELN_ASYNC_B128(src + c, &stage[wid][c]);
        }
    }
    ELN_WAIT_ASYNC();

    // ---- Stage 2: read back, accumulate single-pass statistics -------------
    // chunks [0,32)  : l=0 scalars  (floats [0,128))
    // chunks [32,80) : l=1 vectors  (floats [128,320))
    // chunks [80,120): l=2          (floats [320,480))
    float4 v[4];
    float s0 = 0.f, q0 = 0.f, q1 = 0.f, q2 = 0.f;
#pragma unroll
    for (int j = 0; j < 4; ++j) {
        const int c = lane + 32 * j;
        if (c < ELN_CHUNKS) {
            const float4 t = stage[wid][c];
            v[j] = t;
            const float ss = t.x*t.x + t.y*t.y + t.z*t.z + t.w*t.w;
            if (c < 32)      { s0 += t.x + t.y + t.z + t.w; q0 += ss; }
            else if (c < 80) { q1 += ss; }
            else             { q2 += ss; }
        }
    }

    // ---- Stage 3: wave32 butterfly reduction -------------------------------
#pragma unroll
    for (int m = 16; m >= 1; m >>= 1) {
        s0 += __shfl_xor(s0, m, 32);
        q0 += __shfl_xor(q0, m, 32);
        q1 += __shfl_xor(q1, m, 32);
        q2 += __shfl_xor(q2, m, 32);
    }

    const float mean0 = s0 * (1.0f / 128.0f);
    // var of centered scalars = E[x^2] - mean^2  (clamp tiny negative fp noise)
    const float var0  = fmaxf(q0 * (1.0f / 128.0f) - mean0 * mean0, 0.0f);
    const float inv0  = rsqrtf(var0 + ELN_EPS);
    const float inv1  = rsqrtf(q1 * (1.0f / 192.0f) + ELN_EPS);
    const float inv2  = rsqrtf(q2 * (1.0f / 160.0f) + ELN_EPS);

    // ---- Stage 4: apply + store (coalesced b128) ---------------------------
#pragma unroll
    for (int j = 0; j < 4; ++j) {
        const int c = lane + 32 * j;
        if (c >= ELN_CHUNKS) continue;
        const float4 t = v[j];
        const float4 w = *(const float4*)&wexp[4 * c];
        float4 o;
        if (c < 32) {
            const float4 b = *(const float4*)&bexp[4 * c];
            o.x = (t.x - mean0) * inv0 * w.x + b.x;
            o.y = (t.y - mean0) * inv0 * w.y + b.y;
            o.z = (t.z - mean0) * inv0 * w.z + b.z;
            o.w = (t.w - mean0) * inv0 * w.w + b.w;
        } else {
            const float s = (c < 80) ? inv1 : inv2;
            o.x = t.x * s * w.x;
            o.y = t.y * s * w.y;
            o.z = t.z * s * w.z;
            o.w = t.w * s * w.w;
        }
        dst[c] = o;
    }
}

extern "C" void kernel_launch(void* const* d_in, const int* in_sizes, int n_in,
                              void* d_out, int out_size, void* d_ws, size_t ws_size,
                              hipStream_t stream)
{
    const float* node_input    = (const float*)d_in[0];
    const float* affine_weight = (const float*)d_in[1];
    const float* affine_bias   = (const float*)d_in[2];
    float*       out           = (float*)d_out;

    const int nNodes = in_sizes[0] / ELN_DIM;   // 100000
    const int grid   = (nNodes + WAVES_PER_BLK - 1) / WAVES_PER_BLK;

    hipLaunchKernelGGL(eln_kernel, dim3(grid), dim3(256), 0, stream,
                       node_input, affine_weight, affine_bias, out, nNodes);
}